// BaselineModelSimple_57732950393207
// MI455X (gfx1250) — compile-verified
//
#include <hip/hip_runtime.h>
#include <hip/hip_bf16.h>
#include <stdint.h>

typedef __attribute__((ext_vector_type(2))) float v2f;
typedef __attribute__((ext_vector_type(8))) float v8f;

#define HID 8
#define L1N 1000
#define L1NP 1008   // padded to 63*16
#define OUTC 34
#define BATCH 16

// ---------------- generic zero fill ----------------
__global__ void fill_zero_kernel(unsigned int* __restrict__ p, int n) {
    int i = blockIdx.x * blockDim.x + threadIdx.x;
    if (i < n) p[i] = 0u;
}

// ---------------- graph setup ----------------
__global__ void deg_kernel(const int* __restrict__ row, const int* __restrict__ col,
                           unsigned int* __restrict__ deg, int E) {
    int e = blockIdx.x * blockDim.x + threadIdx.x;
    if (e >= E) return;
    int r = row[e], c = col[e];
    if (r != c) atomicAdd(&deg[r], 1u);
}

__global__ void dinv_kernel(const unsigned int* __restrict__ deg,
                            float* __restrict__ dinv, int N) {
    int n = blockIdx.x * blockDim.x + threadIdx.x;
    if (n >= N) return;
    unsigned int d = deg[n];
    dinv[n] = (d > 0u) ? rsqrtf((float)d) : 0.0f;
}

// count incoming non-zero-weight edges per destination node
__global__ void count_kernel(const int* __restrict__ row, const int* __restrict__ col,
                             const float* __restrict__ dinv,
                             unsigned int* __restrict__ cnt, int E) {
    int e = blockIdx.x * blockDim.x + threadIdx.x;
    if (e >= E) return;
    int r = row[e], c = col[e];
    if (r == c) return;
    if (dinv[r] == 0.0f || dinv[c] == 0.0f) return;
    atomicAdd(&cnt[c], 1u);
}

// ---------------- exclusive prefix sum (3 kernels) ----------------
__global__ __launch_bounds__(256) void scan_block_kernel(
    const unsigned int* __restrict__ cnt, unsigned int* __restrict__ excl,
    unsigned int* __restrict__ blksum, int N) {
    __shared__ unsigned int sh[256];
    int i = blockIdx.x * 256 + threadIdx.x;
    unsigned int v = (i < N) ? cnt[i] : 0u;
    sh[threadIdx.x] = v;
    __syncthreads();
    unsigned int incl = v;
    for (int ofs = 1; ofs < 256; ofs <<= 1) {
        unsigned int add = (threadIdx.x >= ofs) ? sh[threadIdx.x - ofs] : 0u;
        __syncthreads();
        incl += add;
        sh[threadIdx.x] = incl;
        __syncthreads();
    }
    if (i < N) excl[i] = incl - v;
    if (threadIdx.x == 255) blksum[blockIdx.x] = incl;
}

__global__ __launch_bounds__(512) void scan_sums_kernel(unsigned int* __restrict__ blksum,
                                                        int NB) {
    __shared__ unsigned int sh[512];
    int t = threadIdx.x;
    unsigned int v = (t < NB) ? blksum[t] : 0u;
    sh[t] = v;
    __syncthreads();
    unsigned int incl = v;
    for (int ofs = 1; ofs < 512; ofs <<= 1) {
        unsigned int add = (t >= ofs) ? sh[t - ofs] : 0u;
        __syncthreads();
        incl += add;
        sh[t] = incl;
        __syncthreads();
    }
    if (t < NB) blksum[t] = incl - v;
}

__global__ __launch_bounds__(256) void scan_add_kernel(
    unsigned int* __restrict__ excl, const unsigned int* __restrict__ blksum,
    unsigned int* __restrict__ cursor, int N) {
    int i = blockIdx.x * 256 + threadIdx.x;
    if (i >= N) return;
    unsigned int s = excl[i] + blksum[blockIdx.x];
    excl[i] = s;
    cursor[i] = s;
}

// fill CSR (by destination): esrc = source node, ew = edge weight (-dinv[r]*dinv[c])
__global__ void place_kernel(const int* __restrict__ row, const int* __restrict__ col,
                             const float* __restrict__ dinv,
                             unsigned int* __restrict__ cursor,
                             int* __restrict__ esrc, float* __restrict__ ew, int E) {
    int e = blockIdx.x * blockDim.x + threadIdx.x;
    if (e >= E) return;
    int r = row[e], c = col[e];
    if (r == c) return;
    float w = -dinv[r] * dinv[c];
    if (w == 0.0f) return;
    unsigned int slot = atomicAdd(&cursor[c], 1u);
    esrc[slot] = r;
    ew[slot] = w;
}

// ---------------- layer-1 (scalar features), gather-fused ----------------
// acc[n][j] = x[n] * W1[0][0][j]
__global__ void cheb1_first_kernel(const float* __restrict__ x, const float* __restrict__ Wk,
                                   float* __restrict__ acc, int N) {
    int n = blockIdx.x * blockDim.x + threadIdx.x;
    if (n >= N) return;
    float v = x[n];
    float* a = acc + (size_t)n * HID;
#pragma unroll
    for (int j = 0; j < HID; ++j) a[j] = v * Wk[j];
}

// lhat = gather(tprev) + diag*tprev ; tnew = isK1 ? lhat : 2*lhat - tprev2 ; acc += tnew*Wk
__global__ void cheb1_step_kernel(const float* __restrict__ dinv,
                                  const unsigned int* __restrict__ start,
                                  const unsigned int* __restrict__ cnt,
                                  const int* __restrict__ esrc, const float* __restrict__ ew,
                                  const float* __restrict__ tprev, const float* tprev2,
                                  float* tnew, float* __restrict__ acc,
                                  const float* __restrict__ Wk, int N, int isK1) {
    int n = blockIdx.x * blockDim.x + threadIdx.x;
    if (n >= N) return;
    unsigned int b = start[n], e = b + cnt[n];
    float s = 0.0f;
    for (unsigned int i = b; i < e; ++i) s += ew[i] * tprev[esrc[i]];
    float dg = (dinv[n] == 0.0f) ? -1.0f : 0.0f;
    float lh = s + dg * tprev[n];
    float t = isK1 ? lh : (2.0f * lh - tprev2[n]);
    tnew[n] = t;
    float* a = acc + (size_t)n * HID;
#pragma unroll
    for (int j = 0; j < HID; ++j) a[j] += t * Wk[j];
}

__global__ void relu_bias8_kernel(float* __restrict__ h, const float* __restrict__ b, int N) {
    int n = blockIdx.x * blockDim.x + threadIdx.x;
    if (n >= N) return;
    float* p = h + (size_t)n * HID;
#pragma unroll
    for (int j = 0; j < HID; ++j) {
        float v = p[j] + b[j];
        p[j] = v > 0.0f ? v : 0.0f;
    }
}

// ---------------- layer-2 (width-8 features), gather-fused ----------------
// acc2[n][:] = h1[n][:] @ Wk (8x8)
__global__ __launch_bounds__(256) void cheb2_init_kernel(const float* __restrict__ h1,
                                                         const float* __restrict__ Wk,
                                                         float* __restrict__ acc, int N) {
    __shared__ float w[64];
    if (threadIdx.x < 64) w[threadIdx.x] = Wk[threadIdx.x];
    __syncthreads();
    int n = blockIdx.x * blockDim.x + threadIdx.x;
    if (n >= N) return;
    float t[HID];
    const float* hp = h1 + (size_t)n * HID;
#pragma unroll
    for (int j = 0; j < HID; ++j) t[j] = hp[j];
    float* a = acc + (size_t)n * HID;
#pragma unroll
    for (int jj = 0; jj < HID; ++jj) {
        float s = 0.0f;
#pragma unroll
        for (int j = 0; j < HID; ++j) s += t[j] * w[j * HID + jj];
        a[jj] = s;
    }
}

__global__ __launch_bounds__(256) void cheb2_step_kernel(
    const float* __restrict__ dinv,
    const unsigned int* __restrict__ start, const unsigned int* __restrict__ cnt,
    const int* __restrict__ esrc, const float* __restrict__ ew,
    const float* tprev, const float* tprev2, float* tnew,
    float* __restrict__ acc, const float* __restrict__ Wk, int N, int isK1) {
    __shared__ float w[64];
    if (threadIdx.x < 64) w[threadIdx.x] = Wk[threadIdx.x];
    __syncthreads();
    int n = blockIdx.x * blockDim.x + threadIdx.x;
    if (n >= N) return;
    unsigned int b = start[n], e = b + cnt[n];
    float g[HID];
#pragma unroll
    for (int j = 0; j < HID; ++j) g[j] = 0.0f;
    for (unsigned int i = b; i < e; ++i) {
        float wt = ew[i];
        const float4* tr = (const float4*)(tprev + (size_t)esrc[i] * HID);
        float4 lo = tr[0], hi = tr[1];
        g[0] += wt * lo.x; g[1] += wt * lo.y; g[2] += wt * lo.z; g[3] += wt * lo.w;
        g[4] += wt * hi.x; g[5] += wt * hi.y; g[6] += wt * hi.z; g[7] += wt * hi.w;
    }
    float dg = (dinv[n] == 0.0f) ? -1.0f : 0.0f;
    const float* tp = tprev + (size_t)n * HID;
    const float* tq = tprev2 + (size_t)n * HID;
    float t[HID];
#pragma unroll
    for (int j = 0; j < HID; ++j) {
        float lh = g[j] + dg * tp[j];
        t[j] = isK1 ? lh : (2.0f * lh - tq[j]);
    }
    float* tn = tnew + (size_t)n * HID;
#pragma unroll
    for (int j = 0; j < HID; ++j) tn[j] = t[j];
    float* a = acc + (size_t)n * HID;
#pragma unroll
    for (int jj = 0; jj < HID; ++jj) {
        float s = a[jj];
#pragma unroll
        for (int j = 0; j < HID; ++j) s += t[j] * w[j * HID + jj];
        a[jj] = s;
    }
}

// ---------------- lin1 via f32 WMMA (16x16x4), split-K ----------------
__global__ __launch_bounds__(32) void lin1_wmma_kernel(
    const float* __restrict__ H, const float* __restrict__ W,
    float* __restrict__ part, int Ktot, int kchunk_sz) {
    const int lane = threadIdx.x;          // wave32
    const int n0 = blockIdx.x * 16;        // N tile
    const int kbase = blockIdx.y * kchunk_sz;
    const int m = lane & 15;
    const int koff = (lane >> 4) << 1;     // 0 or 2 (K pair per half-wave)
    const int ncol = n0 + m;
    const bool nvalid = (ncol < L1N);
    const int ncl = nvalid ? ncol : 0;

    v8f c = {0.f, 0.f, 0.f, 0.f, 0.f, 0.f, 0.f, 0.f};
    const float* hrow = H + (size_t)m * Ktot + kbase + koff;
    const float* wb = W + (size_t)(kbase + koff) * L1N + ncl;

    for (int kk = 0; kk < kchunk_sz; kk += 4) {
        // A frag: 16x4 f32, lane holds row m, K = koff + {0,1}
        v2f a = *(const v2f*)(hrow + kk);
        // B frag: 4x16 f32, lane holds col ncol, rows K = koff + {0,1}
        float b0 = wb[(size_t)kk * L1N];
        float b1 = wb[(size_t)(kk + 1) * L1N];
        v2f b;
        b.x = nvalid ? b0 : 0.0f;
        b.y = nvalid ? b1 : 0.0f;
        c = __builtin_amdgcn_wmma_f32_16x16x4_f32(
                false, a, false, b, (short)0, c, false, false);
    }
    // C layout: VGPR v -> (M = (lane>=16?8:0)+v, N = lane&15)
    const int cn = n0 + (lane & 15);
    const int mbase = (lane >> 4) * 8;
#pragma unroll
    for (int v = 0; v < 8; ++v)
        atomicAdd(&part[(size_t)(mbase + v) * L1NP + cn], c[v]);
}

__global__ void y1_kernel(const float* __restrict__ part, const float* __restrict__ b,
                          float* __restrict__ y1) {
    int i = blockIdx.x * blockDim.x + threadIdx.x;
    if (i >= BATCH * L1NP) return;
    int n = i % L1NP;
    float v = 0.0f;
    if (n < L1N) {
        v = part[i] + b[n];
        v = v > 0.0f ? v : 0.0f;
    }
    y1[i] = v;
}

__global__ void lin2_kernel(const float* __restrict__ y1, const float* __restrict__ W,
                            const float* __restrict__ b, float* __restrict__ out) {
    int t = blockIdx.x * blockDim.x + threadIdx.x;
    if (t >= BATCH * OUTC) return;
    int bi = t / OUTC, o = t % OUTC;
    const float* yr = y1 + (size_t)bi * L1NP;
    float s = b[o];
    for (int n = 0; n < L1N; ++n) s += yr[n] * W[(size_t)n * OUTC + o];
    out[t] = s;
}

// ---------------- host ----------------
static inline size_t align256(size_t x) { return (x + 255) & ~(size_t)255; }

extern "C" void kernel_launch(void* const* d_in, const int* in_sizes, int n_in,
                              void* d_out, int out_size, void* d_ws, size_t ws_size,
                              hipStream_t stream) {
    const float* x   = (const float*)d_in[0];
    const int*   ei  = (const int*)d_in[1];
    const float* W1  = (const float*)d_in[4];   // (5,1,8)
    const float* b1  = (const float*)d_in[5];
    const float* W2  = (const float*)d_in[6];   // (5,8,8)
    const float* b2  = (const float*)d_in[7];
    const float* l1w = (const float*)d_in[8];   // (65536,1000)
    const float* l1b = (const float*)d_in[9];
    const float* l2w = (const float*)d_in[10];  // (1000,34)
    const float* l2b = (const float*)d_in[11];
    float* out = (float*)d_out;

    const int N = in_sizes[0];
    const int E = in_sizes[1] / 2;
    const int Ktot = N * HID / BATCH;           // 65536
    const int* row = ei;
    const int* col = ei + E;
    const int NB = (N + 255) / 256;             // 512 prefix-sum blocks

    // workspace layout
    char* ws = (char*)d_ws;
    size_t off = 0;
    auto alloc = [&](size_t bytes) { void* p = ws + off; off += align256(bytes); return p; };
    unsigned int* deg    = (unsigned int*)alloc((size_t)N * 4);
    float*        dinv   = (float*)alloc((size_t)N * 4);
    unsigned int* cnt    = (unsigned int*)alloc((size_t)N * 4);
    unsigned int* start  = (unsigned int*)alloc((size_t)N * 4);
    unsigned int* cursor = (unsigned int*)alloc((size_t)N * 4);
    unsigned int* blksum = (unsigned int*)alloc((size_t)NB * 4);
    int*          esrc   = (int*)alloc((size_t)E * 4);
    float*        ew     = (float*)alloc((size_t)E * 4);
    float*        tA     = (float*)alloc((size_t)N * 4);
    float*        tB     = (float*)alloc((size_t)N * 4);
    float*        h1     = (float*)alloc((size_t)N * HID * 4);   // layer1 acc -> h1
    float*        B1     = (float*)alloc((size_t)N * HID * 4);
    float*        B2     = (float*)alloc((size_t)N * HID * 4);
    float*        acc2   = (float*)alloc((size_t)N * HID * 4);   // layer2 acc -> h2
    float*        part   = (float*)alloc((size_t)BATCH * L1NP * 4);
    float*        y1     = (float*)alloc((size_t)BATCH * L1NP * 4);
    (void)ws_size;

    const int TB = 256;
    dim3 blkN((N + TB - 1) / TB), blkE((E + TB - 1) / TB);

    // ---- graph setup: degree, 1/sqrt(deg) ----
    fill_zero_kernel<<<blkN, TB, 0, stream>>>(deg, N);
    deg_kernel<<<blkE, TB, 0, stream>>>(row, col, deg, E);
    dinv_kernel<<<blkN, TB, 0, stream>>>(deg, dinv, N);

    // ---- CSR build by destination (turns all Cheb passes into gathers) ----
    fill_zero_kernel<<<blkN, TB, 0, stream>>>(cnt, N);
    count_kernel<<<blkE, TB, 0, stream>>>(row, col, dinv, cnt, E);
    scan_block_kernel<<<NB, 256, 0, stream>>>(cnt, start, blksum, N);
    scan_sums_kernel<<<1, 512, 0, stream>>>(blksum, NB);
    scan_add_kernel<<<NB, 256, 0, stream>>>(start, blksum, cursor, N);
    place_kernel<<<blkE, TB, 0, stream>>>(row, col, dinv, cursor, esrc, ew, E);

    // ---- ChebConv layer 1 (feature width 1), fused gather+recurrence ----
    cheb1_first_kernel<<<blkN, TB, 0, stream>>>(x, W1 + 0 * HID, h1, N);
    cheb1_step_kernel<<<blkN, TB, 0, stream>>>(dinv, start, cnt, esrc, ew,
                                               x, x, tA, h1, W1 + 1 * HID, N, 1);
    cheb1_step_kernel<<<blkN, TB, 0, stream>>>(dinv, start, cnt, esrc, ew,
                                               tA, x, tB, h1, W1 + 2 * HID, N, 0);
    cheb1_step_kernel<<<blkN, TB, 0, stream>>>(dinv, start, cnt, esrc, ew,
                                               tB, tA, tA, h1, W1 + 3 * HID, N, 0);
    cheb1_step_kernel<<<blkN, TB, 0, stream>>>(dinv, start, cnt, esrc, ew,
                                               tA, tB, tB, h1, W1 + 4 * HID, N, 0);
    relu_bias8_kernel<<<blkN, TB, 0, stream>>>(h1, b1, N);

    // ---- ChebConv layer 2 (feature width 8), fused gather+recurrence ----
    cheb2_init_kernel<<<blkN, TB, 0, stream>>>(h1, W2 + 0 * 64, acc2, N);
    cheb2_step_kernel<<<blkN, TB, 0, stream>>>(dinv, start, cnt, esrc, ew,
                                               h1, h1, B1, acc2, W2 + 1 * 64, N, 1);
    cheb2_step_kernel<<<blkN, TB, 0, stream>>>(dinv, start, cnt, esrc, ew,
                                               B1, h1, B2, acc2, W2 + 2 * 64, N, 0);
    cheb2_step_kernel<<<blkN, TB, 0, stream>>>(dinv, start, cnt, esrc, ew,
                                               B2, B1, B1, acc2, W2 + 3 * 64, N, 0);
    cheb2_step_kernel<<<blkN, TB, 0, stream>>>(dinv, start, cnt, esrc, ew,
                                               B1, B2, B2, acc2, W2 + 4 * 64, N, 0);
    relu_bias8_kernel<<<blkN, TB, 0, stream>>>(acc2, b2, N);   // acc2 == h2 (16 x 65536)

    // ---- lin1 via WMMA f32, split-K into 64 chunks ----
    fill_zero_kernel<<<(BATCH * L1NP + TB - 1) / TB, TB, 0, stream>>>(
        (unsigned int*)part, BATCH * L1NP);
    const int NKCH = 64;
    const int kchunk = Ktot / NKCH;            // 1024
    dim3 g1(L1NP / 16, NKCH);                  // 63 x 64 waves
    lin1_wmma_kernel<<<g1, 32, 0, stream>>>(acc2, l1w, part, Ktot, kchunk);
    y1_kernel<<<(BATCH * L1NP + TB - 1) / TB, TB, 0, stream>>>(part, l1b, y1);

    // ---- lin2 ----
    lin2_kernel<<<(BATCH * OUTC + TB - 1) / TB, TB, 0, stream>>>(y1, l2w, l2b, out);
}